// UV_Aggregator_79044578115814
// MI455X (gfx1250) — compile-verified
//
#include <hip/hip_runtime.h>
#include <hip/hip_bf16.h>
#include <math.h>

typedef __attribute__((ext_vector_type(2))) float v2f;
typedef __attribute__((ext_vector_type(8))) float v8f;

#define BN   128   // nodes
#define HH   30    // history length
#define LL   80    // review length
#define DD   64    // embed dim
#define WD   300   // word dim
#define NF   100   // filters per size
#define XC   432   // concat row stride: 64 + 64 + 300 = 428, padded to 432
#define LROWS 84
#define LSTRIDE 302

// ---------------------------------------------------------------------------
// Kernel A: gather e_uv (v2e[history_uv]) and e_r (r2e[history_r]) into the
// concat buffer columns [0,64) and [64,128); zero the pad cols [428,432).
// ---------------------------------------------------------------------------
__global__ __launch_bounds__(128) void gather_kernel(
    const int* __restrict__ history_uv, const int* __restrict__ history_r,
    const float* __restrict__ v2e, const float* __restrict__ r2e,
    float* __restrict__ xcat) {
  const int rh = blockIdx.x;              // b*H + h
  const int t  = threadIdx.x;             // 0..127
  float* xr = xcat + (size_t)rh * XC;
  if (t < 64) {
    xr[t] = v2e[(size_t)history_uv[rh] * DD + t];
  } else {
    xr[t] = r2e[(size_t)history_r[rh] * DD + (t - 64)];
  }
  if (t < 4) xr[428 + t] = 0.0f;
}

// ---------------------------------------------------------------------------
// Per-(ksize, n-tile) WMMA job.  K is compile-time; Wc/bc are direct kernel
// args so the compiler keeps them in the global address space (global_load,
// SGPR base).  Filter index is clamped (not guarded) so the inner loop has
// no EXEC divergence: lanes n>=100 compute garbage columns that are never
// stored (WMMA D column n depends only on B column n).
// ---------------------------------------------------------------------------
template <int K>
__device__ __forceinline__ void conv_job(
    const float* __restrict__ Wc, const float* __restrict__ bc,
    const float* smem, float* __restrict__ xcat_row,
    int nt, int lane, int sOff) {
  const int lhalf = lane >> 4;     // 0: k pair {0,1}/M 0..7 ; 1: {2,3}/M 8..15
  const int l16   = lane & 15;
  const int n     = nt * 16 + l16;           // filter this lane covers
  const int nc    = (n < NF) ? n : (NF - 1); // clamped -> always-valid loads

  v8f acc0 = {0, 0, 0, 0, 0, 0, 0, 0};
  v8f acc[5];
#pragma unroll
  for (int mt = 0; mt < 5; ++mt) acc[mt] = acc0;

#pragma unroll
  for (int kk = 0; kk < K; ++kk) {
    const float* __restrict__ wrow = Wc + kk * (WD * NF);
    const float* abase0 = smem + (l16 + kk) * LSTRIDE + (lhalf << 1);
    int off = (lhalf << 1) * NF + nc;        // element (c0, nc) of this tap
    for (int cb = 0; cb < WD; cb += 4) {
      v2f bfrag;
      bfrag.x = wrow[off];                   // B[k=c0  ][n]
      bfrag.y = wrow[off + NF];              // B[k=c0+1][n]
      off += 4 * NF;
      const float* abase = abase0 + cb;      // 8B aligned (even col, even stride)
#pragma unroll
      for (int mt = 0; mt < 5; ++mt) {
        v2f afrag = *(const v2f*)(abase + mt * 16 * LSTRIDE);
        acc[mt] = __builtin_amdgcn_wmma_f32_16x16x4_f32(
            false, afrag, false, bfrag, (short)0, acc[mt], false, false);
      }
    }
  }

  // bias + relu + max over valid output positions t in [0, L-K]
  const float bval   = bc[nc];
  const int   Tvalid = LL - K + 1;
  float best = 0.0f;                         // relu output >= 0
#pragma unroll
  for (int mt = 0; mt < 5; ++mt) {
#pragma unroll
    for (int i = 0; i < 8; ++i) {
      const int t = mt * 16 + i + (lhalf << 3);
      const float v = fmaxf(acc[mt][i] + bval, 0.0f);
      if (t < Tvalid) best = fmaxf(best, v);
    }
  }
  best = fmaxf(best, __shfl_xor(best, 16, 32));
  if (lhalf == 0 && n < NF) {
    xcat_row[128 + sOff + n] = best;
  }
}

// ---------------------------------------------------------------------------
// Kernel B: TextCNN via V_WMMA_F32_16X16X4_F32.
// One workgroup per review (b,h).  LDS holds the gathered [84 x 300] word
// matrix (rows 80..83 zero).  8 waves share 21 jobs = (ksize in {3,4,5}) x
// (7 n-tiles of 16 filters); s is wave-uniform so the dispatch is a scalar
// branch and each instantiation sees a compile-time K and a direct global
// weight pointer.
// ---------------------------------------------------------------------------
__global__ __launch_bounds__(256) void conv_kernel(
    const int* __restrict__ history_w, const float* __restrict__ word_emb,
    const float* __restrict__ Wc3, const float* __restrict__ bc3,
    const float* __restrict__ Wc4, const float* __restrict__ bc4,
    const float* __restrict__ Wc5, const float* __restrict__ bc5,
    float* __restrict__ xcat) {
  __shared__ __align__(16) float smem[LROWS * LSTRIDE];
  __shared__ int widx[LL];

  const int rh  = blockIdx.x;
  const int tid = threadIdx.x;

  if (tid < LL) widx[tid] = history_w[(size_t)rh * LL + tid];
  __syncthreads();

  // Gather word embeddings into LDS; rows 80..83 zeroed (safe tap overrun).
  for (int idx = tid; idx < LROWS * WD; idx += 256) {
    const int row = idx / WD;
    const int col = idx - row * WD;
    float v = 0.0f;
    if (row < LL) v = word_emb[(size_t)widx[row] * WD + col];
    smem[row * LSTRIDE + col] = v;
  }
  __syncthreads();

  const int wv   = tid >> 5;
  const int lane = tid & 31;
  float* xr = xcat + (size_t)rh * XC;

  for (int job = wv; job < 21; job += 8) {
    const int s  = job / 7;          // wave-uniform: scalar branch
    const int nt = job - s * 7;
    if (s == 0)      conv_job<3>(Wc3, bc3, smem, xr, nt, lane, 0);
    else if (s == 1) conv_job<4>(Wc4, bc4, smem, xr, nt, lane, NF);
    else             conv_job<5>(Wc5, bc5, smem, xr, nt, lane, 2 * NF);
  }
}

// ---------------------------------------------------------------------------
// Kernel C: 2-layer MLP + attention + softmax + weighted sum.  One block per
// node b, 64 threads = one output dim each.  ~0.3 GFLOP total -> plain VALU.
// ---------------------------------------------------------------------------
__global__ __launch_bounds__(64) void tail_kernel(
    const int* __restrict__ nodes, const float* __restrict__ u2e,
    const float* __restrict__ xcat,
    const float* __restrict__ W_r1, const float* __restrict__ b_r1,
    const float* __restrict__ W_r2, const float* __restrict__ b_r2,
    const float* __restrict__ A1,  const float* __restrict__ ab1,
    const float* __restrict__ A2,  const float* __restrict__ ab2,
    const float* __restrict__ A3,  const float* __restrict__ ab3,
    float* __restrict__ out) {
  const int b = blockIdx.x;
  const int d = threadIdx.x;   // 0..63

  __shared__ float row[XC];
  __shared__ float xbuf[DD];
  __shared__ float obuf[HH][DD];
  __shared__ float a1buf[DD];
  __shared__ float a2buf[DD];
  __shared__ float sbuf[HH];
  __shared__ float ubuf[DD];

  ubuf[d] = u2e[(size_t)nodes[b] * DD + d];
  __syncthreads();

  for (int h = 0; h < HH; ++h) {
    const float* xr = xcat + (size_t)(b * HH + h) * XC;
    for (int k = d; k < 428; k += 64) row[k] = xr[k];
    __syncthreads();

    // x = relu(xcat @ W_r1 + b_r1)
    float acc = b_r1[d];
    for (int k = 0; k < 428; ++k) acc = fmaf(row[k], W_r1[k * DD + d], acc);
    xbuf[d] = fmaxf(acc, 0.0f);
    __syncthreads();

    // o = relu(x @ W_r2 + b_r2)
    acc = b_r2[d];
    for (int k = 0; k < DD; ++k) acc = fmaf(xbuf[k], W_r2[k * DD + d], acc);
    obuf[h][d] = fmaxf(acc, 0.0f);
    __syncthreads();

    // a1 = relu(concat(o, u) @ A1 + ab1)
    acc = ab1[d];
    for (int k = 0; k < DD; ++k) acc = fmaf(obuf[h][k], A1[k * DD + d], acc);
    for (int k = 0; k < DD; ++k) acc = fmaf(ubuf[k], A1[(DD + k) * DD + d], acc);
    a1buf[d] = fmaxf(acc, 0.0f);
    __syncthreads();

    // a2 = relu(a1 @ A2 + ab2)
    acc = ab2[d];
    for (int k = 0; k < DD; ++k) acc = fmaf(a1buf[k], A2[k * DD + d], acc);
    a2buf[d] = fmaxf(acc, 0.0f);
    __syncthreads();

    // s = a2 @ A3 + ab3  (redundantly computed by every thread, thread 0 stores)
    float s = ab3[0];
    for (int k = 0; k < DD; ++k) s = fmaf(a2buf[k], A3[k], s);
    if (d == 0) sbuf[h] = s;
    __syncthreads();
  }

  // softmax over history, then weighted sum of o
  float m = -1e30f;
  for (int h = 0; h < HH; ++h) m = fmaxf(m, sbuf[h]);
  float denom = 0.0f;
  for (int h = 0; h < HH; ++h) denom += expf(sbuf[h] - m);
  float res = 0.0f;
  for (int h = 0; h < HH; ++h) res += obuf[h][d] * (expf(sbuf[h] - m) / denom);
  out[(size_t)b * DD + d] = res;
}

// ---------------------------------------------------------------------------
extern "C" void kernel_launch(void* const* d_in, const int* in_sizes, int n_in,
                              void* d_out, int out_size, void* d_ws, size_t ws_size,
                              hipStream_t stream) {
  const int*   nodes      = (const int*)  d_in[0];
  const int*   history_uv = (const int*)  d_in[1];
  const int*   history_r  = (const int*)  d_in[2];
  const int*   history_w  = (const int*)  d_in[3];
  const float* v2e        = (const float*)d_in[4];
  const float* u2e        = (const float*)d_in[5];
  const float* r2e        = (const float*)d_in[6];
  const float* word_emb   = (const float*)d_in[7];
  const float* Wc3        = (const float*)d_in[8];
  const float* bc3        = (const float*)d_in[9];
  const float* Wc4        = (const float*)d_in[10];
  const float* bc4        = (const float*)d_in[11];
  const float* Wc5        = (const float*)d_in[12];
  const float* bc5        = (const float*)d_in[13];
  const float* W_r1       = (const float*)d_in[14];
  const float* b_r1       = (const float*)d_in[15];
  const float* W_r2       = (const float*)d_in[16];
  const float* b_r2       = (const float*)d_in[17];
  const float* A1         = (const float*)d_in[18];
  const float* ab1        = (const float*)d_in[19];
  const float* A2         = (const float*)d_in[20];
  const float* ab2        = (const float*)d_in[21];
  const float* A3         = (const float*)d_in[22];
  const float* ab3        = (const float*)d_in[23];

  float* out  = (float*)d_out;
  float* xcat = (float*)d_ws;     // [B*H, 432] f32 = 6.64 MB scratch

  const int nrev = BN * HH;       // 3840
  gather_kernel<<<nrev, 128, 0, stream>>>(history_uv, history_r, v2e, r2e, xcat);
  conv_kernel<<<nrev, 256, 0, stream>>>(history_w, word_emb,
                                        Wc3, bc3, Wc4, bc4, Wc5, bc5, xcat);
  tail_kernel<<<BN, 64, 0, stream>>>(nodes, u2e, xcat,
                                     W_r1, b_r1, W_r2, b_r2,
                                     A1, ab1, A2, ab2, A3, ab3, out);
}